// LinearConv2D_14431090114822
// MI455X (gfx1250) — compile-verified
//
#include <hip/hip_runtime.h>

typedef __attribute__((ext_vector_type(2))) float v2f;
typedef __attribute__((ext_vector_type(8))) float v8f;

#define Bn 2
#define Cn 64
#define On 64
#define En 128
#define Tn 512
#define Wn 7
#define TOUT 256
#define FOUT 64

// ---------------------------------------------------------------------------
// Kernel 1: per-(c,f) sums and sums-of-squares of the unfolded x, per w tap.
// Sx[c,f,w] = sum_{b,t} x[b,c,f,2t+w-3]   (zero-padded outside [0,512))
// Qx[c,f,w] = sum_{b,t} x[...]^2
// ---------------------------------------------------------------------------
__global__ void lc_stats1(const float* __restrict__ x,
                          float* __restrict__ Sx, float* __restrict__ Qx) {
  int cf = blockIdx.x;            // c*128 + f
  int c = cf >> 7, f = cf & 127;
  float s[7], q[7];
#pragma unroll
  for (int w = 0; w < 7; ++w) { s[w] = 0.f; q[w] = 0.f; }
  for (int item = threadIdx.x; item < Bn * TOUT; item += blockDim.x) {
    int b = item >> 8, t = item & 255;
    const float* xr = x + (((size_t)b * Cn + c) * En + f) * Tn;
    int tau0 = 2 * t - 3;
#pragma unroll
    for (int w = 0; w < 7; ++w) {
      int tau = tau0 + w;
      if (tau >= 0 && tau < Tn) {
        float v = xr[tau];
        s[w] += v; q[w] += v * v;
      }
    }
  }
  __shared__ float red[8][14];
  int lane = threadIdx.x & 31, wv = threadIdx.x >> 5;
#pragma unroll
  for (int k = 0; k < 14; ++k) {
    float v = (k < 7) ? s[k] : q[k - 7];
    for (int off = 16; off > 0; off >>= 1) v += __shfl_down(v, off, 32);
    if (lane == 0) red[wv][k] = v;
  }
  __syncthreads();
  if (threadIdx.x < 14) {
    float tot = 0.f;
#pragma unroll
    for (int i = 0; i < 8; ++i) tot += red[i][threadIdx.x];
    if (threadIdx.x < 7) Sx[cf * 7 + threadIdx.x] = tot;
    else                 Qx[cf * 7 + threadIdx.x - 7] = tot;
  }
}

// ---------------------------------------------------------------------------
// Kernel 2: per-output-channel BN affine constants.
//   mean[o] = (1/cnt) sum_{d,f,w} W[o,d,f,w]   * Sx[4(o/4)+d, f, w]
//   E[y^2]  = (1/cnt) sum_{d,f,w} W[o,d,f,w]^2 * Qx[4(o/4)+d, f, w]
//   a[o] = gamma*rsqrt(var+eps);  b[o] = 56*(beta - mean*a)
// ---------------------------------------------------------------------------
__global__ void lc_stats2(const float* __restrict__ wgt,
                          const float* __restrict__ gamma,
                          const float* __restrict__ beta,
                          const float* __restrict__ Sx,
                          const float* __restrict__ Qx,
                          float* __restrict__ aO, float* __restrict__ bO) {
  int o = blockIdx.x;
  int cbase = o & ~3;
  float n1 = 0.f, n2 = 0.f;
  for (int idx = threadIdx.x; idx < 4 * En * Wn; idx += blockDim.x) {
    int d = idx / (En * Wn);
    int rem = idx - d * (En * Wn);       // rem = f*7 + w
    float wt = wgt[(size_t)(o * 4 + d) * (En * Wn) + rem];
    int c = cbase + d;
    float sv = Sx[(size_t)c * (En * Wn) + rem];
    float qv = Qx[(size_t)c * (En * Wn) + rem];
    n1 += wt * sv;
    n2 += wt * wt * qv;
  }
  __shared__ float red[4][2];
  int lane = threadIdx.x & 31, wv = threadIdx.x >> 5;
  for (int off = 16; off > 0; off >>= 1) {
    n1 += __shfl_down(n1, off, 32);
    n2 += __shfl_down(n2, off, 32);
  }
  if (lane == 0) { red[wv][0] = n1; red[wv][1] = n2; }
  __syncthreads();
  if (threadIdx.x == 0) {
    float t1 = 0.f, t2 = 0.f;
#pragma unroll
    for (int i = 0; i < 4; ++i) { t1 += red[i][0]; t2 += red[i][1]; }
    const float invcnt = 1.0f / 1835008.0f;  // B*D*E*t*W = 2*4*128*256*7
    float mean = t1 * invcnt;
    float var  = t2 * invcnt - mean * mean;
    float inv  = rsqrtf(var + 1e-5f);
    float a    = gamma[o] * inv;
    aO[o] = a;
    bO[o] = 56.0f * (beta[o] - mean * a);
  }
}

// ---------------------------------------------------------------------------
// Kernel 3: grouped strided conv via V_WMMA_F32_16X16X4_F32.
// One wave per 16(o) x 16(t) output tile at fixed (b, fo).
//  - Stage the x slice (16 ch x 2 f x 37 tau) into LDS with coalesced,
//    contiguous global loads; zero-padding folded into the staging.
//  - 56 WMMAs per tile: 2 f-planes x 7 taps x 4 K-chunks (block-diagonal A).
//    B operands come from LDS as aligned 8-byte ds_load_b64 (channel-inner
//    layout, pad 18 -> conflict-free banking, even-channel 8B alignment).
//  - Epilogue applies the BN affine + LeakyReLU.
// ---------------------------------------------------------------------------
__global__ void __launch_bounds__(32)
lc_wmma_main(const float* __restrict__ x, const float* __restrict__ wgt,
             const float* __restrict__ aO, const float* __restrict__ bO,
             float* __restrict__ out) {
  int id = blockIdx.x;
  int tt = id & 15;
  int fo = (id >> 4) & 63;
  int ob = (id >> 10) & 3;
  int b  = id >> 12;
  int t0 = tt << 4, o0 = ob << 4, c0 = o0, f0 = fo << 1;

  int L  = threadIdx.x;
  int m  = L & 15;          // A: row M; B/D: column N (= local t)
  int hi = L >> 4;          // 0: K=0,1 / M=r ; 1: K=2,3 / M=r+8
  int d0 = hi << 1;         // lane's K pair = depth pair (d0, d0+1)
  int mq = m >> 2;          // which K-chunk carries this lane's A row

  // LDS tile: xs[f][tau_local][channel]; 18-pad => even-channel addresses are
  // 8B aligned and read pattern (stride 36 floats across lanes) is
  // conflict-free on 64 banks.
  __shared__ float xs[2][40][18];

  // Stage: 2*16*37 = 1184 elements, 37 uniform iterations of the wave.
  // Contiguous 37-float runs per (f, ch) -> coalesced HBM access.
  for (int idx = L; idx < 1184; idx += 32) {
    int fi = idx / 592;                 // 592 = 16*37
    int r  = idx - fi * 592;
    int ch = r / 37;
    int j  = r - ch * 37;
    int tau = 2 * t0 - 3 + j;
    float v = 0.f;
    if (tau >= 0 && tau < Tn)
      v = x[(((size_t)b * Cn + (c0 + ch)) * En + (f0 + fi)) * Tn + tau];
    xs[fi][j][ch] = v;
  }

  // Preload this lane's 28 weights: (2 f) x (7 w) x (d0, d0+1)
  float wa[2][7], wb[2][7];
#pragma unroll
  for (int fi = 0; fi < 2; ++fi) {
    const float* wp = wgt + (((size_t)(o0 + m) * 4 + d0) * En + (f0 + fi)) * Wn;
#pragma unroll
    for (int w = 0; w < 7; ++w) {
      wa[fi][w] = wp[w];
      wb[fi][w] = wp[En * Wn + w];   // next d: stride E*W = 896
    }
  }

  __syncthreads();

  v8f acc = {};
#pragma unroll
  for (int fi = 0; fi < 2; ++fi) {
#pragma unroll
    for (int w = 0; w < 7; ++w) {
      int j = 2 * m + w;              // tau_local = p - (2*t0 - 3), 0..36
#pragma unroll
      for (int q = 0; q < 4; ++q) {
        v2f bv = *(const v2f*)&xs[fi][j][(q << 2) + d0];  // ds_load_b64
        float sel = (mq == q) ? 1.f : 0.f;                // block-diagonal A
        v2f av = { sel * wa[fi][w], sel * wb[fi][w] };
        acc = __builtin_amdgcn_wmma_f32_16x16x4_f32(
            /*neg_a=*/false, av, /*neg_b=*/false, bv,
            /*c_mod=*/(short)0, acc, /*reuse_a=*/false, /*reuse_b=*/false);
      }
    }
  }

  // Epilogue: D VGPR r -> element (M = r + 8*hi, N = m)
#pragma unroll
  for (int r = 0; r < 8; ++r) {
    int o = o0 + r + (hi << 3);
    float v = aO[o] * acc[r] + bO[o];
    v = (v >= 0.f) ? v : 0.01f * v;
    out[(((size_t)b * On + o) * FOUT + fo) * TOUT + t0 + m] = v;
  }
}

extern "C" void kernel_launch(void* const* d_in, const int* in_sizes, int n_in,
                              void* d_out, int out_size, void* d_ws, size_t ws_size,
                              hipStream_t stream) {
  (void)in_sizes; (void)n_in; (void)out_size; (void)ws_size;
  const float* x     = (const float*)d_in[0];
  const float* wgt   = (const float*)d_in[1];
  const float* gamma = (const float*)d_in[2];
  const float* beta  = (const float*)d_in[3];
  float* out = (float*)d_out;

  float* ws = (float*)d_ws;
  float* Sx = ws;                 // 64*128*7 = 57344 floats
  float* Qx = ws + 57344;         // 57344 floats
  float* aO = ws + 114688;        // 64 floats
  float* bO = ws + 114752;        // 64 floats

  lc_stats1<<<Cn * En, 256, 0, stream>>>(x, Sx, Qx);
  lc_stats2<<<On, 128, 0, stream>>>(wgt, gamma, beta, Sx, Qx, aO, bO);
  lc_wmma_main<<<8192, 32, 0, stream>>>(x, wgt, aO, bO, out);
}